// Weighted3DTransformLoss_23467701306049
// MI455X (gfx1250) — compile-verified
//
#include <hip/hip_runtime.h>

typedef float v2f __attribute__((ext_vector_type(2)));
typedef float v8f __attribute__((ext_vector_type(8)));

#define BB      32
#define KK      8
#define HH      240
#define WW      320
#define HWPX    (HH * WW)         // 76800 pixels per (b, channel)
#define NTILES  (HWPX / 16)       // 4800 16-pixel tiles per batch
#define CHUNKS  20                // blocks per batch
#define NWAVES  8                 // waves per block (256 threads, wave32)
#define TSTRIDE (CHUNKS * NWAVES) // 160
#define NITERS  (NTILES / TSTRIDE) // exactly 30 tiles per wave

// D = T(3x32) * Q(32x16) per tile via V_WMMA_F32_16X16X4_F32, K walked in 8
// steps of 4 (one transform k per step, j' = 0..3). A 9th WMMA with A = -I
// (rows 0..2) and B = target folds in the subtraction, leaving err in D.
__global__ __launch_bounds__(256)
void w3dtl_main(const float* __restrict__ pts,
                const float* __restrict__ masks,
                const float* __restrict__ tfms,
                const float* __restrict__ tgt,
                const float* __restrict__ numpts,
                float* __restrict__ partial)
{
    const int tid   = threadIdx.x;
    const int w     = tid >> 5;
    const int lane  = tid & 31;
    const int n16   = lane & 15;   // pixel-in-tile (N) / output row (M) index
    const int half  = lane >> 4;   // which K-half this lane's VGPRs hold
    const int b     = blockIdx.x / CHUNKS;
    const int chunk = blockIdx.x % CHUNKS;

    const float* ptsB   = pts   + (size_t)b * 3  * HWPX;
    const float* masksB = masks + (size_t)b * KK * HWPX;
    const float* tgtB   = tgt   + (size_t)b * 3  * HWPX;
    const float* tfmsB  = tfms  + (size_t)b * KK * 12;

    // WMMA f32 16x16x4 register layout: lane L -> M (or N) = L%16,
    // VGPR v holds K = 2*(L/16) + v.
    const int i  = n16;        // A-matrix row (output component; rows 3..15 zero)
    const int j0 = half * 2;   // K index held in VGPR 0 of this lane
    const int j1 = j0 + 1;     // K index held in VGPR 1

    // A matrices: 8 transform slices + 1 (-identity) slice. Constant per wave.
    v2f a[9];
#pragma unroll
    for (int s = 0; s < KK; ++s) {
        float a0 = 0.f, a1 = 0.f;
        if (i < 3) {
            const float* T = tfmsB + (s * 3 + i) * 4;  // tfms[b,s,i,0..3]
            a0 = T[j0];
            a1 = T[j1];
        }
        a[s] = v2f{a0, a1};
    }
    a[8] = v2f{ (i == j0) ? -1.f : 0.f,
                (i == j1) ? -1.f : 0.f };

    float acc = 0.f;
    const int t0 = chunk * NWAVES + w;

#pragma unroll 1
    for (int it = 0; it < NITERS; ++it) {
        const int t = t0 + it * TSTRIDE;
        const int n = t * 16 + n16;

        // Uniform (scalar-branch) prefetch of the next tile this wave touches.
        if (it + 1 < NITERS) {
            __builtin_prefetch(&masksB[n + 16 * TSTRIDE], 0, 1);
            __builtin_prefetch(&ptsB  [n + 16 * TSTRIDE], 0, 1);
        }

        // B-matrix point factors for this lane's two K rows:
        // lanes 0-15 -> (p_x, p_y); lanes 16-31 -> (p_z, 1).
        const float pA  = ptsB[j0 * HWPX + n];                     // j0 in {0,2}
        const float pB  = (half == 0) ? ptsB[HWPX + n] : 1.0f;     // j1 in {1,3}
        const float tv0 = tgtB[j0 * HWPX + n];
        const float tv1 = (half == 0) ? tgtB[HWPX + n] : 0.0f;

        v8f c = {0.f, 0.f, 0.f, 0.f, 0.f, 0.f, 0.f, 0.f};
#pragma unroll
        for (int s = 0; s < KK; ++s) {
            const float m = masksB[s * HWPX + n];
            v2f bm = v2f{m * pA, m * pB};
            c = __builtin_amdgcn_wmma_f32_16x16x4_f32(
                    false, a[s], false, bm, (short)0, c, false, false);
        }
        // err = pred - target via A = -I slice
        v2f bt = v2f{tv0, tv1};
        c = __builtin_amdgcn_wmma_f32_16x16x4_f32(
                false, a[8], false, bt, (short)0, c, false, false);

        // Rows 0..2 (lanes 0-15, VGPR 0..2) hold err; all other rows are 0.
#pragma unroll
        for (int r = 0; r < 8; ++r) acc += c[r] * c[r];
    }

    // wave32 reduction
#pragma unroll
    for (int off = 16; off > 0; off >>= 1)
        acc += __shfl_xor(acc, off, 32);

    __shared__ float wpart[NWAVES];
    if (lane == 0) wpart[w] = acc;
    __syncthreads();
    if (tid == 0) {
        float s = 0.f;
#pragma unroll
        for (int x = 0; x < NWAVES; ++x) s += wpart[x];
        partial[blockIdx.x] = s / numpts[b];   // per-batch normalization
    }
}

// Deterministic final reduction of BB*CHUNKS partials.
__global__ __launch_bounds__(256)
void w3dtl_reduce(const float* __restrict__ partial, int n, float* __restrict__ out)
{
    __shared__ float sm[256];
    float s = 0.f;
    for (int idx = threadIdx.x; idx < n; idx += 256) s += partial[idx];
    sm[threadIdx.x] = s;
    __syncthreads();
    for (int off = 128; off > 0; off >>= 1) {
        if (threadIdx.x < off) sm[threadIdx.x] += sm[threadIdx.x + off];
        __syncthreads();
    }
    if (threadIdx.x == 0) out[0] = 0.5f * sm[0];
}

extern "C" void kernel_launch(void* const* d_in, const int* in_sizes, int n_in,
                              void* d_out, int out_size, void* d_ws, size_t ws_size,
                              hipStream_t stream)
{
    const float* inputpts   = (const float*)d_in[0];
    const float* inputmasks = (const float*)d_in[1];
    const float* inputtfms  = (const float*)d_in[2];
    const float* targetpts  = (const float*)d_in[3];
    const float* numpts     = (const float*)d_in[4];
    float*       out        = (float*)d_out;
    float*       partial    = (float*)d_ws;       // BB*CHUNKS floats = 2560 B

    const int nblocks = BB * CHUNKS;              // 640
    w3dtl_main<<<nblocks, 256, 0, stream>>>(inputpts, inputmasks, inputtfms,
                                            targetpts, numpts, partial);
    w3dtl_reduce<<<1, 256, 0, stream>>>(partial, nblocks, out);
}